// Decoder_2267742732493
// MI455X (gfx1250) — compile-verified
//
#include <hip/hip_runtime.h>
#include <math.h>

// ---------------------------------------------------------------------------
// CDNA5 WMMA (wave32): V_WMMA_F32_16X16X4_F32
//   A: 16x4 f32  -> 2 VGPRs/lane   (lane%16 = M row; lane/16 selects K pair)
//   B: 4x16 f32  -> 2 VGPRs/lane   (lane%16 = N col; lane/16 selects K pair)
//   C/D: 16x16   -> 8 VGPRs/lane   (lane%16 = N col; VGPR v -> M = v + (lane/16)*8)
// ---------------------------------------------------------------------------
typedef __attribute__((ext_vector_type(2))) float v2f;
typedef __attribute__((ext_vector_type(8))) float v8f;

#define BLK 256

static inline int gdim(long n, int blk) { return (int)((n + blk - 1) / blk); }

// --------------------------------- kernels ---------------------------------

__global__ void k_zero(float* __restrict__ p, long n) {
  long i = (long)blockIdx.x * blockDim.x + threadIdx.x;
  if (i < n) p[i] = 0.0f;
}

__global__ void k_deg(const int* __restrict__ row, int E, float* __restrict__ deg) {
  int i = blockIdx.x * blockDim.x + threadIdx.x;
  if (i < E) atomicAdd(&deg[row[i]], 1.0f);
}

__global__ void k_dis(const float* __restrict__ deg, float* __restrict__ dis, int N) {
  int i = blockIdx.x * blockDim.x + threadIdx.x;
  if (i < N) {
    float d = deg[i];
    dis[i] = (d > 0.0f) ? rsqrtf(fmaxf(d, 1.0f)) : 0.0f;
  }
}

__global__ void k_norm(const int* __restrict__ row, const int* __restrict__ col,
                       const float* __restrict__ dis, float* __restrict__ nrm, int E) {
  int i = blockIdx.x * blockDim.x + threadIdx.x;
  if (i < E) nrm[i] = -dis[row[i]] * dis[col[i]];
}

// Sparse unpool (float4 channels): out[b,r,:] = sum_{j=0..2} val[3r+j]*in[b,col[3r+j],:]
__global__ void k_pool4(const float4* __restrict__ in, float4* __restrict__ out,
                        const int* __restrict__ cols, const float* __restrict__ vals,
                        int B, int Nout, int Nin, int C4) {
  long total = (long)B * Nout * C4;
  long i = (long)blockIdx.x * blockDim.x + threadIdx.x;
  if (i >= total) return;
  int c = (int)(i % C4);
  long t = i / C4;
  int r = (int)(t % Nout);
  int b = (int)(t / Nout);
  const float4* xb = in + (long)b * Nin * C4;
  int base = 3 * r;
  float v0 = vals[base + 0], v1 = vals[base + 1], v2 = vals[base + 2];
  float4 a0 = xb[(long)cols[base + 0] * C4 + c];
  float4 a1 = xb[(long)cols[base + 1] * C4 + c];
  float4 a2 = xb[(long)cols[base + 2] * C4 + c];
  float4 acc;
  acc.x = v0 * a0.x + v1 * a1.x + v2 * a2.x;
  acc.y = v0 * a0.y + v1 * a1.y + v2 * a2.y;
  acc.z = v0 * a0.z + v1 * a1.z + v2 * a2.z;
  acc.w = v0 * a0.w + v1 * a1.w + v2 * a2.w;
  out[(long)b * Nout * C4 + (long)r * C4 + c] = acc;
}

// Normalized-Laplacian matvec scatter (float4 gather, scalar f32 atomics):
// y[b,row[e],:] += nrm[e]*x[b,col[e],:]
__global__ void k_prop4(const float4* __restrict__ x, float* __restrict__ y,
                        const int* __restrict__ row, const int* __restrict__ col,
                        const float* __restrict__ nrm, int B, int N, int C4, int E) {
  long total = (long)B * E;
  long i = (long)blockIdx.x * blockDim.x + threadIdx.x;
  if (i >= total) return;
  int e = (int)(i % E);
  int b = (int)(i / E);
  float w = nrm[e];
  const float4* xs = x + ((long)b * N + col[e]) * C4;
  float* yd = y + ((long)b * N + row[e]) * C4 * 4;
  for (int c = 0; c < C4; ++c) {
    float4 v = xs[c];
    atomicAdd(&yd[4 * c + 0], w * v.x);
    atomicAdd(&yd[4 * c + 1], w * v.y);
    atomicAdd(&yd[4 * c + 2], w * v.z);
    atomicAdd(&yd[4 * c + 3], w * v.w);
  }
}

// Chebyshev recurrence combine: t2 = 2*t2 - t0
__global__ void k_combine(float* __restrict__ t2, const float* __restrict__ t0, long n) {
  long i = (long)blockIdx.x * blockDim.x + threadIdx.x;
  if (i < n) t2[i] = 2.0f * t2[i] - t0[i];
}

__global__ void k_bias_elu(float* __restrict__ p, const float* __restrict__ bias,
                           long n, int C) {
  long i = (long)blockIdx.x * blockDim.x + threadIdx.x;
  if (i >= n) return;
  float v = p[i] + bias[i % C];
  p[i] = (v > 0.0f) ? v : (expf(v) - 1.0f);
}

__global__ void k_bias(float* __restrict__ p, const float* __restrict__ bias,
                       long n, int C) {
  long i = (long)blockIdx.x * blockDim.x + threadIdx.x;
  if (i < n) p[i] += bias[i % C];
}

// Zero-pad final cheb weights: W[6,32,3] -> WP[6,32,16]
__global__ void k_padw(const float* __restrict__ W, float* __restrict__ WP) {
  int i = blockIdx.x * blockDim.x + threadIdx.x;   // 6*32*16 = 3072
  if (i >= 6 * 32 * 16) return;
  int c = i & 15;
  int r = i >> 4;                                  // k*32 + cin
  WP[i] = (c < 3) ? W[r * 3 + c] : 0.0f;
}

// Final epilogue: d_out[m,0..2] = OUTP[m, 0..2] + bias[c]
__global__ void k_final_out(const float* __restrict__ outp, const float* __restrict__ bias,
                            float* __restrict__ dst, long M) {
  long i = (long)blockIdx.x * blockDim.x + threadIdx.x;
  if (i >= M * 3) return;
  long m = i / 3;
  int c = (int)(i % 3);
  dst[i] = outp[m * 16 + c] + bias[c];
}

// ---------------------------------------------------------------------------
// Branch-free WMMA GEMM, one wave computes a 16 x (NT*16) row strip:
//   C[M, NT*16] += A[M,K] @ Bm[K, NT*16]   (row-major f32; M%16==0, K%4==0)
// A tile loaded once per K-step (global_load_b64) and reused across NT WMMAs.
// ---------------------------------------------------------------------------
template <int NT>
__global__ void k_gemm_nt(const float* __restrict__ A, const float* __restrict__ Bm,
                          float* __restrict__ C, int M, int K) {
  const int Nf = NT * 16;
  int wave = blockIdx.x * (blockDim.x >> 5) + (threadIdx.x >> 5);
  int mTiles = M >> 4;
  if (wave >= mTiles) return;                 // wave-uniform
  int lane = threadIdx.x & 31;
  int half = lane >> 4;
  int lcol = lane & 15;
  int mBase = wave << 4;

  v8f acc[NT];
#pragma unroll
  for (int nt = 0; nt < NT; ++nt)
#pragma unroll
    for (int v = 0; v < 8; ++v)
      acc[nt][v] = C[(long)(mBase + v + half * 8) * Nf + nt * 16 + lcol];

  long arow = (long)(mBase + lcol) * K;
  for (int k0 = 0; k0 < K; k0 += 4) {
    int ka = k0 + half * 2;
    v2f a;
    a[0] = A[arow + ka];
    a[1] = A[arow + ka + 1];
#pragma unroll
    for (int nt = 0; nt < NT; ++nt) {
      v2f b;
      b[0] = Bm[(long)ka * Nf + nt * 16 + lcol];
      b[1] = Bm[(long)(ka + 1) * Nf + nt * 16 + lcol];
      acc[nt] = __builtin_amdgcn_wmma_f32_16x16x4_f32(false, a, false, b, (short)0,
                                                      acc[nt], false, false);
    }
  }

#pragma unroll
  for (int nt = 0; nt < NT; ++nt)
#pragma unroll
    for (int v = 0; v < 8; ++v)
      C[(long)(mBase + v + half * 8) * Nf + nt * 16 + lcol] = acc[nt][v];
}

// Generic single-tile variant for wide N (latent linear, Nfull % 16 == 0).
__global__ void k_gemm_gen(const float* __restrict__ A, const float* __restrict__ Bm,
                           float* __restrict__ C, int M, int K, int Nfull, int Ntiles) {
  int wave = blockIdx.x * (blockDim.x >> 5) + (threadIdx.x >> 5);
  int mTiles = M >> 4;
  long total = (long)mTiles * Ntiles;
  if (wave >= total) return;
  int mt = wave / Ntiles;
  int nt = wave % Ntiles;
  int lane = threadIdx.x & 31;
  int half = lane >> 4;
  int lcol = lane & 15;
  int mBase = mt << 4;
  int n = (nt << 4) + lcol;

  v8f c;
#pragma unroll
  for (int v = 0; v < 8; ++v)
    c[v] = C[(long)(mBase + v + half * 8) * Nfull + n];

  long arow = (long)(mBase + lcol) * K;
  for (int k0 = 0; k0 < K; k0 += 4) {
    int ka = k0 + half * 2;
    v2f a, b;
    a[0] = A[arow + ka];
    a[1] = A[arow + ka + 1];
    b[0] = Bm[(long)ka * Nfull + n];
    b[1] = Bm[(long)(ka + 1) * Nfull + n];
    c = __builtin_amdgcn_wmma_f32_16x16x4_f32(false, a, false, b, (short)0, c,
                                              false, false);
  }

#pragma unroll
  for (int v = 0; v < 8; ++v)
    C[(long)(mBase + v + half * 8) * Nfull + n] = c[v];
}

// ------------------------------ host sequencing -----------------------------

static void launch_gemm_nt(hipStream_t s, const float* A, const float* Bm, float* C,
                           long M, int K, int NT) {
  int blocks = (int)((M / 16 + 7) / 8);            // 8 waves per 256-thread block
  switch (NT) {
    case 1: k_gemm_nt<1><<<blocks, BLK, 0, s>>>(A, Bm, C, (int)M, K); break;
    case 2: k_gemm_nt<2><<<blocks, BLK, 0, s>>>(A, Bm, C, (int)M, K); break;
    default: k_gemm_nt<4><<<blocks, BLK, 0, s>>>(A, Bm, C, (int)M, K); break;
  }
}

// One ChebConv (K=6), sym-normalized Laplacian, lambda_max=2.
// Cout must be a multiple of 16 here (final layer is padded by the caller).
// epilogue: 0 = bias+ELU, 1 = none.
static void run_cheb(hipStream_t s, float* tx0, float* tx1, float* tx2, float* outb,
                     const float* W, const float* bias,
                     const int* erow, const int* ecol,
                     float* DEG, float* DIS, float* NORM,
                     int B, int N, int Cin, int Cout, int epilogue) {
  const int E = 6 * N;
  const long M = (long)B * N;
  const long elemsIn = M * Cin;
  const long elemsOut = M * Cout;
  const int C4 = Cin / 4;
  const int NT = Cout / 16;

  // edge normalization
  k_zero<<<gdim(N, BLK), BLK, 0, s>>>(DEG, N);
  k_deg<<<gdim(E, BLK), BLK, 0, s>>>(erow, E, DEG);
  k_dis<<<gdim(N, BLK), BLK, 0, s>>>(DEG, DIS, N);
  k_norm<<<gdim(E, BLK), BLK, 0, s>>>(erow, ecol, DIS, NORM, E);

  // out = Tx0 @ W0
  k_zero<<<gdim(elemsOut, BLK), BLK, 0, s>>>(outb, elemsOut);
  launch_gemm_nt(s, tx0, W, outb, M, Cin, NT);

  // Tx1 = L_hat x ; out += Tx1 @ W1
  k_zero<<<gdim(elemsIn, BLK), BLK, 0, s>>>(tx1, elemsIn);
  k_prop4<<<gdim((long)B * E, BLK), BLK, 0, s>>>((const float4*)tx0, tx1, erow, ecol,
                                                 NORM, B, N, C4, E);
  launch_gemm_nt(s, tx1, W + (size_t)Cin * Cout, outb, M, Cin, NT);

  for (int k = 2; k < 6; ++k) {
    k_zero<<<gdim(elemsIn, BLK), BLK, 0, s>>>(tx2, elemsIn);
    k_prop4<<<gdim((long)B * E, BLK), BLK, 0, s>>>((const float4*)tx1, tx2, erow, ecol,
                                                   NORM, B, N, C4, E);
    k_combine<<<gdim(elemsIn, BLK), BLK, 0, s>>>(tx2, tx0, elemsIn);
    launch_gemm_nt(s, tx2, W + (size_t)k * Cin * Cout, outb, M, Cin, NT);
    float* t = tx0; tx0 = tx1; tx1 = tx2; tx2 = t;
  }

  if (epilogue == 0)
    k_bias_elu<<<gdim(elemsOut, BLK), BLK, 0, s>>>(outb, bias, elemsOut, Cout);
}

extern "C" void kernel_launch(void* const* d_in, const int* in_sizes, int n_in,
                              void* d_out, int out_size, void* d_ws, size_t ws_size,
                              hipStream_t stream) {
  (void)in_sizes; (void)n_in; (void)out_size; (void)ws_size;

  // setup_inputs() dict order (lists flattened):
  const float* x     = (const float*)d_in[0];
  const float* lin_w = (const float*)d_in[1];
  const float* lin_b = (const float*)d_in[2];
  const float* cheb_w[5]; const float* cheb_b[5];
  for (int i = 0; i < 5; ++i) { cheb_w[i] = (const float*)d_in[3 + i];
                                cheb_b[i] = (const float*)d_in[8 + i]; }
  // up_rows (13..16) unused: structure is repeat(arange(n_out), 3).
  const int*   up_cols[4]; const float* up_vals[4];
  const int*   erow[4];    const int*   ecol[4];
  for (int i = 0; i < 4; ++i) {
    up_cols[i] = (const int*)d_in[17 + i];
    up_vals[i] = (const float*)d_in[21 + i];
    erow[i]    = (const int*)d_in[25 + i];
    ecol[i]    = (const int*)d_in[29 + i];
  }

  // workspace layout (floats)
  const size_t MAXBUF = (size_t)16 * 65536 * 32;  // 33.55M floats
  float* TXA  = (float*)d_ws;
  float* TXB  = TXA + MAXBUF;
  float* TXC  = TXB + MAXBUF;                     // also final padded out [M,16]
  float* OUTB = TXC + MAXBUF;
  float* DEG  = OUTB + MAXBUF;
  float* DIS  = DEG + 65536;
  float* NORM = DIS + 65536;                      // 6*65536 floats
  float* WP   = NORM + 6 * 65536;                 // 6*32*16 padded final weights

  float* out = (float*)d_out;

  const int sizes[5] = {65536, 16384, 4096, 1024, 256};
  const int cin_[4]  = {64, 64, 32, 32};
  const int cout_[4] = {64, 32, 32, 32};

  // 1) latent linear: OUTB[16,16384] = x[16,64] @ lin_w[64,16384] + lin_b
  {
    const long n = 16L * 16384;
    k_zero<<<gdim(n, BLK), BLK, 0, stream>>>(OUTB, n);
    const int Ntiles = 16384 / 16;
    const long tiles = 1L * Ntiles;               // M=16 -> one m-tile
    k_gemm_gen<<<(int)((tiles + 7) / 8), BLK, 0, stream>>>(x, lin_w, OUTB, 16, 64,
                                                           16384, Ntiles);
    k_bias<<<gdim(n, BLK), BLK, 0, stream>>>(OUTB, lin_b, n, 16384);
  }

  // 2) four deblocks, coarse -> fine; OUTB reinterpreted as [16, N, C]
  float* hprev = OUTB;
  for (int j = 0; j < 4; ++j) {
    int t = 3 - j;
    int Nout = sizes[t], Nin = sizes[t + 1];
    long n4 = (long)16 * Nout * (cin_[j] / 4);
    k_pool4<<<gdim(n4, BLK), BLK, 0, stream>>>((const float4*)hprev, (float4*)TXA,
                                               up_cols[t], up_vals[t],
                                               16, Nout, Nin, cin_[j] / 4);
    run_cheb(stream, TXA, TXB, TXC, OUTB, cheb_w[j], cheb_b[j], erow[t], ecol[t],
             DEG, DIS, NORM, 16, Nout, cin_[j], cout_[j], /*epilogue=*/0);
    hprev = OUTB;
  }

  // 3) final ChebConv 32 -> 3 (padded to 16 cols) at finest level
  k_padw<<<gdim(6 * 32 * 16, BLK), BLK, 0, stream>>>(cheb_w[4], WP);
  run_cheb(stream, OUTB, TXA, TXB, TXC, WP, cheb_b[4], erow[0], ecol[0],
           DEG, DIS, NORM, 16, 65536, 32, 16, /*epilogue=*/1);
  const long M = 16L * 65536;
  k_final_out<<<gdim(M * 3, BLK), BLK, 0, stream>>>(TXC, cheb_b[4], out, M);
}